// GNN_MagNet_68908455297560
// MI455X (gfx1250) — compile-verified
//
#include <hip/hip_runtime.h>
#include <hip/hip_bf16.h>

typedef __attribute__((ext_vector_type(2))) float v2f;
typedef __attribute__((ext_vector_type(8))) float v8f;

// theta = +/- 2*pi*0.25 = +/- float32(pi/2); cos is even, sin is odd.
#define COS_C (-4.37113883e-08f)
#define SIN_C (1.0f)

// ---------------------------------------------------------------- utilities
__global__ void fill_kernel(float* __restrict__ p, float v, long n) {
  long i = (long)blockIdx.x * blockDim.x + threadIdx.x;
  if (i < n) p[i] = v;
}

__global__ void relu_kernel(float* __restrict__ p, long n) {
  long i = (long)blockIdx.x * blockDim.x + threadIdx.x;
  if (i < n) p[i] = fmaxf(p[i], 0.0f);
}

// out is N x 16 (padded); cols >= Fout get 0.
__global__ void bias_init_kernel(float* __restrict__ out, const float* __restrict__ b,
                                 long n, int Fout) {
  long i = (long)blockIdx.x * blockDim.x + threadIdx.x;
  if (i >= n) return;
  int f = (int)(i & 15);
  out[i] = (f < Fout) ? b[f] : 0.0f;
}

// ------------------------------------------------------- graph normalization
__global__ void deg_kernel(const int* __restrict__ e0, const int* __restrict__ e1,
                           float* __restrict__ deg, int E) {
  int e = blockIdx.x * blockDim.x + threadIdx.x;
  if (e >= E) return;
  atomicAdd(&deg[e0[e]], 0.5f);
  atomicAdd(&deg[e1[e]], 0.5f);
}

__global__ void dinv_kernel(float* __restrict__ deg, int N) {  // in-place deg -> dinv
  int i = blockIdx.x * blockDim.x + threadIdx.x;
  if (i >= N) return;
  float d = deg[i];
  deg[i] = (d > 0.0f) ? rsqrtf(d) : 0.0f;
}

__global__ void anorm_kernel(const int* __restrict__ e0, const int* __restrict__ e1,
                             const float* __restrict__ dinv, float* __restrict__ a, int E) {
  int e = blockIdx.x * blockDim.x + threadIdx.x;
  if (e >= E) return;
  a[e] = dinv[e0[e]] * 0.5f * dinv[e1[e]];
}

// ------------------------------------------------- fused complex propagation
// t1[dst][ch][f] += n_ch * x_ch[src][f] over all directed edges.
// channels: 0: nr*xr  1: ni*xi  2: ni*xr  3: nr*xi
// x is stored N x F (row stride F = 16 or 32), t is N x 4F.
__global__ void prop1_kernel(const int* __restrict__ e0, const int* __restrict__ e1,
                             const float* __restrict__ an,
                             const float* __restrict__ xr, const float* __restrict__ xi,
                             float* __restrict__ t1, int E, int F) {
  int d = blockIdx.x * blockDim.x + threadIdx.x;
  if (d >= 2 * E) return;
  int fwd = (d < E);
  int e = fwd ? d : d - E;
  int u = e0[e], v = e1[e];
  int src = fwd ? u : v;
  int dst = fwd ? v : u;
  float a = an[e];
  float nr = -a * COS_C;
  float ni = (fwd ? -a : a) * SIN_C;
  const float4* pr = reinterpret_cast<const float4*>(xr + (size_t)src * F);
  const float4* pm = reinterpret_cast<const float4*>(xi + (size_t)src * F);
  float* td = t1 + (size_t)dst * 4 * F;
  int nf4 = F >> 2;
  for (int q = 0; q < nf4; ++q) {
    float4 r = pr[q], m = pm[q];
    float rv[4] = {r.x, r.y, r.z, r.w};
    float mv[4] = {m.x, m.y, m.z, m.w};
    int fb = q * 4;
#pragma unroll
    for (int j = 0; j < 4; ++j) {
      atomicAdd(&td[0 * F + fb + j], nr * rv[j]);
      atomicAdd(&td[1 * F + fb + j], ni * mv[j]);
      atomicAdd(&td[2 * F + fb + j], ni * rv[j]);
      atomicAdd(&td[3 * F + fb + j], nr * mv[j]);
    }
  }
}

// t2 pre-initialized to -t0; here: t2[dst][c][f] += 2*n_c * t1[src][c][f]
__global__ void prop2_kernel(const int* __restrict__ e0, const int* __restrict__ e1,
                             const float* __restrict__ an,
                             const float* __restrict__ t1, float* __restrict__ t2,
                             int E, int F) {
  int d = blockIdx.x * blockDim.x + threadIdx.x;
  if (d >= 2 * E) return;
  int fwd = (d < E);
  int e = fwd ? d : d - E;
  int u = e0[e], v = e1[e];
  int src = fwd ? u : v;
  int dst = fwd ? v : u;
  float a = an[e];
  float nr = -a * COS_C;
  float ni = (fwd ? -a : a) * SIN_C;
  float n2[4] = {2.0f * nr, 2.0f * ni, 2.0f * ni, 2.0f * nr};
  const float4* ps = reinterpret_cast<const float4*>(t1 + (size_t)src * 4 * F);
  float* td = t2 + (size_t)dst * 4 * F;
  int nf4 = F >> 2;
#pragma unroll
  for (int c = 0; c < 4; ++c) {
    float nc = n2[c];
    for (int q = 0; q < nf4; ++q) {
      float4 s = ps[c * nf4 + q];
      float sv[4] = {s.x, s.y, s.z, s.w};
      int fb = c * F + q * 4;
#pragma unroll
      for (int j = 0; j < 4; ++j) atomicAdd(&td[fb + j], nc * sv[j]);
    }
  }
}

// t2 := -t0 where t0 = (xr, xi, xr, xi); x stride = F
__global__ void init_t2_kernel(const float* __restrict__ xr, const float* __restrict__ xi,
                               float* __restrict__ t2, long NF, int F) {
  long i = (long)blockIdx.x * blockDim.x + threadIdx.x;
  if (i >= NF) return;
  long n = i / F;
  int f = (int)(i % F);
  float r = -xr[i], m = -xi[i];
  float* d = t2 + n * 4 * F;
  d[0 * F + f] = r;
  d[1 * F + f] = m;
  d[2 * F + f] = r;
  d[3 * F + f] = m;
}

// ------------------------------------------------------------- WMMA f32 GEMM
// Out[M x 16] += (A0 [+/- A1])[M x FIN] @ W[FIN x Fout], W zero-padded to 16
// cols in LDS. One wave per 16-row C tile; v_wmma_f32_16x16x4_f32 over K.
// ISA f32 layouts: A: row = lane%16, K pair by lane half. B: col = lane%16,
// K pair by lane half. C/D: vgpr j -> row j + 8*(lane>=16), col lane%16.
template <int FIN, bool HAS_A1>
__global__ void gemm_wmma_kernel(const float* __restrict__ A0, const float* __restrict__ A1,
                                 float sgn, int strideA,
                                 const float* __restrict__ W, int Fout,
                                 float* __restrict__ Out, int nTiles) {
  __shared__ float Wl[FIN * 16];
  for (int i = threadIdx.x; i < FIN * 16; i += blockDim.x) {
    int r = i >> 4, col = i & 15;
    Wl[i] = (col < Fout) ? W[r * Fout + col] : 0.0f;
  }
  __syncthreads();

  int wave = (int)((blockIdx.x * blockDim.x + threadIdx.x) >> 5);
  int lane = threadIdx.x & 31;
  if (wave >= nTiles) return;  // uniform per wave; after the barrier
  int half = lane >> 4;
  int l = lane & 15;
  long row0 = (long)wave * 16;
  const float* a0 = A0 + (row0 + l) * (size_t)strideA;
  const float* a1 = A1 + (row0 + l) * (size_t)strideA;
  float* outp = Out + row0 * 16;

  v8f c;
#pragma unroll
  for (int j = 0; j < 8; ++j) c[j] = outp[(j + 8 * half) * 16 + l];

#pragma unroll
  for (int k0 = 0; k0 < FIN; k0 += 4) {
    int ka = k0 + 2 * half;
    v2f a, b;
    if (HAS_A1) {
      a.x = a0[ka] + sgn * a1[ka];
      a.y = a0[ka + 1] + sgn * a1[ka + 1];
    } else {
      a.x = a0[ka];
      a.y = a0[ka + 1];
    }
    b.x = Wl[ka * 16 + l];
    b.y = Wl[(ka + 1) * 16 + l];
    c = __builtin_amdgcn_wmma_f32_16x16x4_f32(false, a, false, b, (short)0, c,
                                              false, false);
  }

#pragma unroll
  for (int j = 0; j < 8; ++j) outp[(j + 8 * half) * 16 + l] = c[j];
}

// ------------------------------------------------------------------- head
// layer-3 outputs are N x 16 padded; real half in cols [0,8), imag buffer same.
__global__ void head_kernel(const float* __restrict__ xr, const float* __restrict__ xi,
                            const float* __restrict__ Wh, const float* __restrict__ bh,
                            float* __restrict__ out, int N) {
  int n = blockIdx.x * blockDim.x + threadIdx.x;
  if (n >= N) return;
  const float* r = xr + (size_t)n * 16;
  const float* m = xi + (size_t)n * 16;
  float s = bh[0];
#pragma unroll
  for (int j = 0; j < 8; ++j) s += r[j] * Wh[j] + m[j] * Wh[8 + j];
  out[n] = s;
}

// ---------------------------------------------------------------- host side
static inline int cdivl(long a, long b) { return (int)((a + b - 1) / b); }

template <int FIN>
static void run_layer(const float* inR, const float* inI,
                      const float* Wl, const float* bl, int Fout,
                      float* outR, float* outI, float* t1, float* t2,
                      const int* e0, const int* e1, const float* an,
                      int N, int E, bool do_relu, hipStream_t s) {
  const int B = 256;
  int nTiles = N / 16;
  int gemmBlocks = cdivl((long)nTiles * 32, B);
  long nOut = (long)N * 16;
  long nT = (long)N * 4 * FIN;
  long nIn = (long)N * FIN;
  size_t wOff = (size_t)FIN * Fout;

  // out = bias (zero-padded to 16 cols)
  bias_init_kernel<<<cdivl(nOut, B), B, 0, s>>>(outR, bl, nOut, Fout);
  bias_init_kernel<<<cdivl(nOut, B), B, 0, s>>>(outI, bl, nOut, Fout);
  // out += t0 @ W[0]
  gemm_wmma_kernel<FIN, false><<<gemmBlocks, B, 0, s>>>(inR, inR, 0.0f, FIN, Wl, Fout,
                                                        outR, nTiles);
  gemm_wmma_kernel<FIN, false><<<gemmBlocks, B, 0, s>>>(inI, inI, 0.0f, FIN, Wl, Fout,
                                                        outI, nTiles);
  // t1 = prop(t0)
  fill_kernel<<<cdivl(nT, B), B, 0, s>>>(t1, 0.0f, nT);
  prop1_kernel<<<cdivl(2L * E, B), B, 0, s>>>(e0, e1, an, inR, inI, t1, E, FIN);
  // out += (t1 combos) @ W[1]
  gemm_wmma_kernel<FIN, true><<<gemmBlocks, B, 0, s>>>(t1 + 0 * FIN, t1 + 1 * FIN, -1.0f,
                                                       4 * FIN, Wl + wOff, Fout, outR, nTiles);
  gemm_wmma_kernel<FIN, true><<<gemmBlocks, B, 0, s>>>(t1 + 2 * FIN, t1 + 3 * FIN, +1.0f,
                                                       4 * FIN, Wl + wOff, Fout, outI, nTiles);
  // t2 = 2*prop(t1) - t0
  init_t2_kernel<<<cdivl(nIn, B), B, 0, s>>>(inR, inI, t2, nIn, FIN);
  prop2_kernel<<<cdivl(2L * E, B), B, 0, s>>>(e0, e1, an, t1, t2, E, FIN);
  // out += (t2 combos) @ W[2]
  gemm_wmma_kernel<FIN, true><<<gemmBlocks, B, 0, s>>>(t2 + 0 * FIN, t2 + 1 * FIN, -1.0f,
                                                       4 * FIN, Wl + 2 * wOff, Fout, outR, nTiles);
  gemm_wmma_kernel<FIN, true><<<gemmBlocks, B, 0, s>>>(t2 + 2 * FIN, t2 + 3 * FIN, +1.0f,
                                                       4 * FIN, Wl + 2 * wOff, Fout, outI, nTiles);
  if (do_relu)
    relu_kernel<<<cdivl(nOut, B), B, 0, s>>>(outR, nOut);
}

extern "C" void kernel_launch(void* const* d_in, const int* in_sizes, int n_in,
                              void* d_out, int out_size, void* d_ws, size_t ws_size,
                              hipStream_t stream) {
  const float* x  = (const float*)d_in[0];
  const int*   ei = (const int*)d_in[1];
  const float* W1 = (const float*)d_in[2];
  const float* b1 = (const float*)d_in[3];
  const float* W2 = (const float*)d_in[4];
  const float* b2 = (const float*)d_in[5];
  const float* W3 = (const float*)d_in[6];
  const float* b3 = (const float*)d_in[7];
  const float* Wh = (const float*)d_in[8];
  const float* bh = (const float*)d_in[9];

  int N = in_sizes[0] / 32;   // 100000
  int E = in_sizes[1] / 2;    // 1600000
  const int* e0 = ei;
  const int* e1 = ei + E;

  float* ws = (float*)d_ws;
  size_t off = 0;
  float* deg = ws + off; off += (size_t)N;            // deg -> dinv (in place)
  float* an  = ws + off; off += (size_t)E;            // per-undirected-edge norm magnitude
  float* AR  = ws + off; off += (size_t)N * 16;
  float* AI  = ws + off; off += (size_t)N * 16;
  float* BR  = ws + off; off += (size_t)N * 16;
  float* BI  = ws + off; off += (size_t)N * 16;
  float* t1  = ws + off; off += (size_t)N * 4 * 32;
  float* t2  = ws + off; off += (size_t)N * 4 * 32;
  (void)ws_size; (void)n_in;

  const int B = 256;
  // --- Laplacian normalization (everything L2-resident) ---
  fill_kernel<<<cdivl(N, B), B, 0, stream>>>(deg, 0.0f, N);
  deg_kernel<<<cdivl(E, B), B, 0, stream>>>(e0, e1, deg, E);
  dinv_kernel<<<cdivl(N, B), B, 0, stream>>>(deg, N);
  anorm_kernel<<<cdivl(E, B), B, 0, stream>>>(e0, e1, deg, an, E);

  // --- three MagNet conv layers ---
  run_layer<32>(x,  x,  W1, b1, 16, AR, AI, t1, t2, e0, e1, an, N, E, true,  stream);
  run_layer<16>(AR, AI, W2, b2, 16, BR, BI, t1, t2, e0, e1, an, N, E, true,  stream);
  run_layer<16>(BR, BI, W3, b3,  8, AR, AI, t1, t2, e0, e1, an, N, E, false, stream);

  // --- linear head on concat([xr, xi]) ---
  head_kernel<<<cdivl(N, B), B, 0, stream>>>(AR, AI, Wh, bh, (float*)d_out, N);
}